// cheb_conv_withSAt_53214644797927
// MI455X (gfx1250) — compile-verified
//
#include <hip/hip_runtime.h>

typedef float v2f __attribute__((ext_vector_type(2)));
typedef float v8f __attribute__((ext_vector_type(8)));

namespace {
constexpr int B_   = 32;
constexpr int N_   = 900;
constexpr int FIN  = 64;
constexpr int FOUT = 64;
constexpr int T_   = 12;
constexpr int K_   = 3;
constexpr int J_   = FOUT * T_;        // 768 fused (o,t) columns
constexpr int NT16 = (N_ + 15) / 16;   // 57 tiles of 16
}

// ---------------------------------------------------------------------------
// Stage 1: y[b,k,m,o,t] = sum_f x[b,m,f,t] * theta[k,f,o]
// One wave per t (12 waves / block); block = (b,k,m-tile). C = 16(m) x 64(o).
// WMMA f32 16x16x4: A[M=m,K=f] from x (f-strided), B[K=f,N=o] from theta.
// Edge rows handled branchlessly: clamped address + 0/1 mask multiply.
// ---------------------------------------------------------------------------
__global__ __launch_bounds__(384)
void s1_xtheta(const float* __restrict__ x, const float* __restrict__ theta,
               float* __restrict__ y, int b0) {
  const int mt = blockIdx.x;           // 0..56
  const int k  = blockIdx.y;           // 0..2
  const int bb = blockIdx.z;           // local batch (workspace index)
  const int b  = b0 + bb;              // global batch
  const int t    = (int)threadIdx.x >> 5;   // wave id == t (0..11)
  const int lane = (int)threadIdx.x & 31;
  const int lrow  = lane & 15;
  const int khalf = (lane >> 4) << 1;  // lanes 0-15: K=0,1 ; lanes 16-31: K=2,3
  const int m0 = mt * 16;
  const int m  = m0 + lrow;
  const bool mok = (m < N_);
  const int mclamp  = mok ? m : (N_ - 1);
  const float mskf  = mok ? 1.0f : 0.0f;

  v8f acc[4] = {};

  // x[b][m][f][t]; stride T_ between consecutive f (always-valid address)
  const float* xb = x + ((size_t)((size_t)b * N_ + mclamp) * FIN) * T_ + t;

  #pragma unroll 4
  for (int f0 = 0; f0 < FIN; f0 += 4) {
    const int fb = f0 + khalf;
    v2f a;
    a.x = xb[(size_t)fb * T_]       * mskf;  // A VGPR0: K=fb
    a.y = xb[(size_t)(fb + 1) * T_] * mskf;  // A VGPR1: K=fb+1
    #pragma unroll
    for (int ot = 0; ot < 4; ++ot) {
      const int o = ot * 16 + lrow;
      const float* th = theta + ((size_t)k * FIN + fb) * FOUT + o;
      v2f bf;
      bf.x = th[0];        // B VGPR0: row K=fb,   col o
      bf.y = th[FOUT];     // B VGPR1: row K=fb+1, col o
      acc[ot] = __builtin_amdgcn_wmma_f32_16x16x4_f32(
          false, a, false, bf, (short)0, acc[ot], false, false);
    }
  }

  // C/D layout: VGPR r -> M = r (+8 for lanes 16-31), N = lane&15
  const int mhi = (lane >> 4) << 3;
  #pragma unroll
  for (int ot = 0; ot < 4; ++ot) {
    const int o = ot * 16 + lrow;
    #pragma unroll
    for (int r = 0; r < 8; ++r) {
      const int mm = m0 + mhi + r;
      if (mm < N_) {
        y[((((size_t)bb * K_ + k) * N_ + mm) * FOUT + o) * T_ + t] = acc[ot][r];
      }
    }
  }
}

// ---------------------------------------------------------------------------
// Stage 2: out[b,n,j] = relu( sum_k sum_m (cheb[k,m,n]*At[b,m,n]) * y[b,k,m,j] )
// Block = (n-tile of 16, b). 8 waves; wave w owns j in [w*96, w*96+96).
// A tile (masked^T, 16n x 16m) built once per 16-m chunk in LDS (stride 18,
// conflict-free ds_load_b64 fragments). B fragments loaded coalesced from
// global y (L2-resident). 900 = 56*16 + 4 -> tail is exactly one K=4 step.
// The K-step loop is split into a fully-unrolled 4-step fast path (uniform
// branch) so the compiler can clause the loads and pipeline the 24 WMMAs.
// ---------------------------------------------------------------------------
__global__ __launch_bounds__(256)
void s2_spatial(const float* __restrict__ cheb, const float* __restrict__ At,
                const float* __restrict__ y, float* __restrict__ out, int b0) {
  const int nt = blockIdx.x;           // 0..56
  const int bb = blockIdx.y;           // local batch
  const int b  = b0 + bb;
  const int n0 = nt * 16;
  const int tid  = (int)threadIdx.x;
  const int wave = tid >> 5;
  const int lane = tid & 31;
  const int lrow  = lane & 15;
  const int khalf = (lane >> 4) << 1;
  const int j0 = wave * 96;

  __shared__ __align__(16) float Alds[16 * 18];  // [n][m], padded stride 18

  v8f acc[6] = {};

  const float* atb = At + (size_t)b * N_ * N_;

  // Branchless fill coordinates (clamped address + 0/1 mask)
  const int mm_f = tid >> 4;           // 0..15 (m within chunk)
  const int nn_f = tid & 15;           // 0..15 (n within tile)
  const int gn_f = n0 + nn_f;
  const int gnc  = (gn_f < N_) ? gn_f : (N_ - 1);
  const float nmask = (gn_f < N_) ? 1.0f : 0.0f;

  for (int k = 0; k < K_; ++k) {
    const float* chb = cheb + (size_t)k * N_ * N_;
    const float* yk  = y + ((size_t)bb * K_ + k) * (size_t)N_ * J_;
    for (int m0 = 0; m0 < N_; m0 += 16) {
      __syncthreads();
      {
        // 256 threads fill 16x16 masked^T tile: Alds[n][m] = cheb[k,m,n]*At[b,m,n]
        const int gm  = m0 + mm_f;
        const int gmc = (gm < N_) ? gm : (N_ - 1);
        const float msk = ((gm < N_) ? 1.0f : 0.0f) * nmask;
        const size_t off = (size_t)gmc * N_ + gnc;
        Alds[nn_f * 18 + mm_f] = chb[off] * atb[off] * msk;
      }
      __syncthreads();

      const float* yrow = yk + (size_t)m0 * J_ + j0 + lrow;
      if (m0 + 16 <= N_) {
        // Fast path: full 16-m chunk = 4 K-steps, fully unrolled.
        #pragma unroll
        for (int s = 0; s < 4; ++s) {
          const int mb = s * 4 + khalf;
          const v2f a = *(const v2f*)&Alds[lrow * 18 + mb];  // ds_load_b64
          #pragma unroll
          for (int jt = 0; jt < 6; ++jt) {
            v2f bf;
            bf.x = yrow[(size_t)mb * J_ + jt * 16];        // row K=mb (coalesced)
            bf.y = yrow[(size_t)(mb + 1) * J_ + jt * 16];  // row K=mb+1
            acc[jt] = __builtin_amdgcn_wmma_f32_16x16x4_f32(
                false, a, false, bf, (short)0, acc[jt], false, false);
          }
        }
      } else {
        // Tail chunk (m0 = 896): exactly 4 valid rows = one K=4 step.
        const int mb = khalf;
        const v2f a = *(const v2f*)&Alds[lrow * 18 + mb];
        #pragma unroll
        for (int jt = 0; jt < 6; ++jt) {
          v2f bf;
          bf.x = yrow[(size_t)mb * J_ + jt * 16];
          bf.y = yrow[(size_t)(mb + 1) * J_ + jt * 16];
          acc[jt] = __builtin_amdgcn_wmma_f32_16x16x4_f32(
              false, a, false, bf, (short)0, acc[jt], false, false);
        }
      }
    }
  }

  // ReLU + store final output: out[b][n][o][t], j = o*T + t
  const int nhi = (lane >> 4) << 3;
  #pragma unroll
  for (int jt = 0; jt < 6; ++jt) {
    const int j = j0 + jt * 16 + lrow;
    const int o = j / T_;
    const int t = j % T_;
    #pragma unroll
    for (int r = 0; r < 8; ++r) {
      const int n = n0 + nhi + r;
      if (n < N_) {
        const float v = acc[jt][r];
        out[(((size_t)b * N_ + n) * FOUT + o) * T_ + t] = v > 0.0f ? v : 0.0f;
      }
    }
  }
}

// ---------------------------------------------------------------------------
extern "C" void kernel_launch(void* const* d_in, const int* in_sizes, int n_in,
                              void* d_out, int out_size, void* d_ws, size_t ws_size,
                              hipStream_t stream) {
  (void)in_sizes; (void)n_in; (void)out_size;
  const float* x     = (const float*)d_in[0];  // (B,N,F_in,T)
  const float* At    = (const float*)d_in[1];  // (B,N,N)
  const float* cheb  = (const float*)d_in[2];  // (K,N,N)
  const float* theta = (const float*)d_in[3];  // (K,F_in,F_out)
  float* out = (float*)d_out;
  float* y   = (float*)d_ws;                   // (nb,K,N,F_out*T) chunked

  const size_t perB = (size_t)K_ * N_ * J_ * sizeof(float);  // 8.3 MB / batch
  int nb = (ws_size >= perB) ? (int)(ws_size / perB) : 1;
  if (nb > B_) nb = B_;
  if (nb < 1) nb = 1;

  for (int b0 = 0; b0 < B_; b0 += nb) {
    const int cur = (B_ - b0 < nb) ? (B_ - b0) : nb;
    dim3 g1(NT16, K_, cur);
    s1_xtheta<<<g1, dim3(384, 1, 1), 0, stream>>>(x, theta, y, b0);
    dim3 g2(NT16, cur, 1);
    s2_spatial<<<g2, dim3(256, 1, 1), 0, stream>>>(cheb, At, y, out, b0);
  }
}